// MyModel_1090921693859
// MI455X (gfx1250) — compile-verified
//
#include <hip/hip_runtime.h>
#include <math.h>

// ---------------- constants (match reference) ----------------
#define NN   20000
#define EE   320000
#define RR   8
#define HH   4
#define CIN  256
#define CHID 256
#define COUT 256
#define OC1  (CHID/HH)   // 64
#define SLOPE_ACT 0.01f
#define SLOPE_ATT 0.2f

typedef __attribute__((ext_vector_type(2))) float v2f;
typedef __attribute__((ext_vector_type(8))) float v8f;

// ---------------- small helpers ----------------
__device__ __forceinline__ void atomicMaxF(float* addr, float val) {
    // sign-split trick: works for any mix of signs, avoids relying on float atomicMax
    if (val >= 0.0f) atomicMax((int*)addr, __float_as_int(val));
    else             atomicMin((unsigned int*)addr, __float_as_uint(val));
}

__global__ void k_fill(float* __restrict__ p, float v, long long n) {
    long long i = (long long)blockIdx.x * blockDim.x + threadIdx.x;
    if (i < n) p[i] = v;
}

__global__ void k_init_bias(float* __restrict__ out, const float* __restrict__ b, int n, int c) {
    long long t = (long long)blockIdx.x * blockDim.x + threadIdx.x;
    if (t >= (long long)n * c) return;
    out[t] = b[t % c];
}

__global__ void k_deg_edges(float* __restrict__ deg, const int* __restrict__ dst, int e) {
    int i = blockIdx.x * blockDim.x + threadIdx.x;
    if (i < e) atomicAdd(&deg[dst[i]], 1.0f);
}

__global__ void k_rsqrt(float* __restrict__ d, int n) {
    int i = blockIdx.x * blockDim.x + threadIdx.x;
    if (i < n) d[i] = rsqrtf(d[i]);   // deg >= 1 always (self loops)
}

// Y[N,O] = X[N,K] @ W[K,O] (small K; plain VALU)
__global__ void k_small_mm(const float* __restrict__ X, const float* __restrict__ W,
                           float* __restrict__ Y, int n, int k, int o) {
    long long t = (long long)blockIdx.x * blockDim.x + threadIdx.x;
    if (t >= (long long)n * o) return;
    int row = (int)(t / o), col = (int)(t % o);
    float s = 0.0f;
    for (int i = 0; i < k; ++i) s += X[(long long)row * k + i] * W[i * o + col];
    Y[t] = s;
}

// g = dis[n]^2 * h + b  (self-loop term of GCN, also initializes accumulator)
__global__ void k_gcn_self(const float* __restrict__ h, const float* __restrict__ dis,
                           const float* __restrict__ b, float* __restrict__ g, int n, int c) {
    long long t = (long long)blockIdx.x * blockDim.x + threadIdx.x;
    if (t >= (long long)n * c) return;
    int row = (int)(t / c);
    float dn = dis[row];
    g[t] = dn * dn * h[t] + b[t % c];
}

__global__ void k_gcn_scatter(const float* __restrict__ h, const float* __restrict__ dis,
                              const int* __restrict__ src, const int* __restrict__ dst,
                              float* __restrict__ g, int e, int c) {
    long long t = (long long)blockIdx.x * blockDim.x + threadIdx.x;
    if (t >= (long long)e * c) return;
    int ed = (int)(t / c), ch = (int)(t % c);
    int s = src[ed], d = dst[ed];
    atomicAdd(&g[(long long)d * c + ch], dis[s] * dis[d] * h[(long long)s * c + ch]);
}

__global__ void k_leaky(float* __restrict__ x, float slope, long long n) {
    long long i = (long long)blockIdx.x * blockDim.x + threadIdx.x;
    if (i < n) { float v = x[i]; x[i] = (v >= 0.0f) ? v : slope * v; }
}

__global__ void k_concat(const float* __restrict__ a, const float* __restrict__ b,
                         float* __restrict__ o, int n) {
    long long t = (long long)blockIdx.x * blockDim.x + threadIdx.x;
    if (t >= (long long)n * 256) return;
    int row = (int)(t / 256), c = (int)(t % 256);
    o[t] = (c < 128) ? a[(long long)row * 128 + c] : b[(long long)row * 128 + (c - 128)];
}

// ---------------- WMMA f32 GEMM (wave32, 1 wave -> 16x64 strip) ----------------
// C[z] = A[M,K] @ B[z][K,Nout] (+ C preload if accumulate) (+ bias[col])
// Each wave computes 4 adjacent 16x16 N-tiles. Per K-step, the A fragment and all
// four B fragments are loaded into DISTINCT registers before any WMMA is issued,
// so the load clause overlaps the 4 independent WMMA accumulator chains instead
// of serializing on a single reused fragment register.
// Layouts per CDNA5 ISA 7.12.2:
//   A 16x4 : lane L holds A[m0 + L%16][k0 + 2*(L/16) + j] in fragment elem j
//   B 4x16 : lane L holds B[k0 + 2*(L/16) + j][n0 + L%16] in fragment elem j
//   C 16x16: lane L elem v holds C[m0 + v + 8*(L/16)][n0 + L%16]
template <int K>
__global__ void k_wmma_gemm(const float* __restrict__ A, const float* __restrict__ B,
                            float* __restrict__ C, const float* __restrict__ bias,
                            int M, int Nout, int accumulate) {
    const int lane = threadIdx.x & 31;
    const int half = lane >> 4;
    const int l16  = lane & 15;
    const int n0 = blockIdx.x * 64;          // 16x64 strip
    const int m0 = blockIdx.y * 16;
    const float* Br = B + (size_t)blockIdx.z * K * Nout;
    float*       Cr = C + (size_t)blockIdx.z * M * Nout;

    v8f acc[4] = {{}, {}, {}, {}};
    if (accumulate) {
#pragma unroll
        for (int j = 0; j < 4; ++j)
#pragma unroll
            for (int v = 0; v < 8; ++v)
                acc[j][v] = Cr[(size_t)(m0 + v + 8 * half) * Nout + n0 + 16 * j + l16];
    }

    const float* Arow = A + (size_t)(m0 + l16) * K;
#pragma unroll 4
    for (int k0 = 0; k0 < K; k0 += 4) {
        const int ka = k0 + 2 * half;
        const float* Bp = Br + (size_t)ka * Nout + n0 + l16;

        // issue ALL loads for this K-step first (distinct registers)
        v2f a, b0, b1, b2, b3;
        a[0]  = Arow[ka];
        a[1]  = Arow[ka + 1];
        b0[0] = Bp[0];        b0[1] = Bp[Nout];
        b1[0] = Bp[16];       b1[1] = Bp[16 + Nout];
        b2[0] = Bp[32];       b2[1] = Bp[32 + Nout];
        b3[0] = Bp[48];       b3[1] = Bp[48 + Nout];

        // then the 4 independent WMMA chains
        acc[0] = __builtin_amdgcn_wmma_f32_16x16x4_f32(false, a, false, b0,
                                                       (short)0, acc[0], false, false);
        acc[1] = __builtin_amdgcn_wmma_f32_16x16x4_f32(false, a, false, b1,
                                                       (short)0, acc[1], false, false);
        acc[2] = __builtin_amdgcn_wmma_f32_16x16x4_f32(false, a, false, b2,
                                                       (short)0, acc[2], false, false);
        acc[3] = __builtin_amdgcn_wmma_f32_16x16x4_f32(false, a, false, b3,
                                                       (short)0, acc[3], false, false);
    }
    if (bias) {
#pragma unroll
        for (int j = 0; j < 4; ++j) {
            float bv = bias[n0 + 16 * j + l16];
#pragma unroll
            for (int v = 0; v < 8; ++v) acc[j][v] += bv;
        }
    }
#pragma unroll
    for (int j = 0; j < 4; ++j)
#pragma unroll
        for (int v = 0; v < 8; ++v)
            Cr[(size_t)(m0 + v + 8 * half) * Nout + n0 + 16 * j + l16] = acc[j][v];
}

// ---------------- RGAT attention pieces ----------------
// qn[r][n][h] = xw[r][n][:] @ q[:,h] ; kn likewise
__global__ void k_qk(const float* __restrict__ xw, const float* __restrict__ q,
                     const float* __restrict__ k, float* __restrict__ qn,
                     float* __restrict__ kn, long long rn, int c, int h) {
    long long t = (long long)blockIdx.x * blockDim.x + threadIdx.x;
    if (t >= rn * h) return;
    long long row = t / h;
    int hh = (int)(t % h);
    const float* xr = xw + row * c;
    float sq = 0.0f, sk = 0.0f;
    for (int i = 0; i < c; ++i) {
        float x = xr[i];
        sq += x * q[i * h + hh];
        sk += x * k[i * h + hh];
    }
    qn[t] = sq;
    kn[t] = sk;
}

__global__ void k_edge_alpha(const float* __restrict__ qn, const float* __restrict__ kn,
                             const int* __restrict__ src, const int* __restrict__ dst,
                             const int* __restrict__ et, float* __restrict__ a,
                             float* __restrict__ amax, int e, int h, int n) {
    long long t = (long long)blockIdx.x * blockDim.x + threadIdx.x;
    if (t >= (long long)e * h) return;
    int ed = (int)(t / h), hh = (int)(t % h);
    int r = et[ed], s = src[ed], d = dst[ed];
    float al = qn[((long long)r * n + d) * h + hh] + kn[((long long)r * n + s) * h + hh];
    al = (al >= 0.0f) ? al : SLOPE_ATT * al;
    a[t] = al;
    atomicMaxF(&amax[(long long)d * h + hh], al);
}

__global__ void k_edge_exp(const float* __restrict__ amax, const int* __restrict__ dst,
                           float* __restrict__ a, float* __restrict__ denom, int e, int h) {
    long long t = (long long)blockIdx.x * blockDim.x + threadIdx.x;
    if (t >= (long long)e * h) return;
    int ed = (int)(t / h), hh = (int)(t % h);
    int d = dst[ed];
    float ex = expf(a[t] - amax[(long long)d * h + hh]);
    a[t] = ex;
    atomicAdd(&denom[(long long)d * h + hh], ex);
}

__global__ void k_edge_norm(const float* __restrict__ denom, const int* __restrict__ dst,
                            float* __restrict__ a, int e, int h) {
    long long t = (long long)blockIdx.x * blockDim.x + threadIdx.x;
    if (t >= (long long)e * h) return;
    int ed = (int)(t / h), hh = (int)(t % h);
    a[t] = a[t] / denom[(long long)dst[ed] * h + hh];
}

// out[dst][c] += alpha[e][c/oc] * xw[et][src][c]
__global__ void k_edge_scatter(const float* __restrict__ xw, const float* __restrict__ a,
                               const int* __restrict__ src, const int* __restrict__ dst,
                               const int* __restrict__ et, float* __restrict__ out,
                               int e, int c, int oc, int h, int n) {
    long long t = (long long)blockIdx.x * blockDim.x + threadIdx.x;
    if (t >= (long long)e * c) return;
    int ed = (int)(t / c), ch = (int)(t % c);
    int r = et[ed], s = src[ed], d = dst[ed];
    float al = a[(long long)ed * h + (ch / oc)];
    float v  = xw[((long long)r * n + s) * c + ch];
    atomicAdd(&out[(long long)d * c + ch], al * v);
}

// ---------------- host-side orchestration ----------------
static inline int cdiv(long long a, int b) { return (int)((a + b - 1) / b); }

static void rgat_layer(const float* X, const float* W, const float* q, const float* k,
                       const float* b, float* xw, float* qn, float* kn,
                       float* amax, float* denom, float* outbuf, float* a_out,
                       const int* src, const int* dst, const int* et,
                       hipStream_t s) {
    // xw[r] = X @ W[r]  (WMMA f32, 16x64 strips)
    dim3 g1(COUT / 64, NN / 16, RR);
    k_wmma_gemm<CIN><<<g1, 32, 0, s>>>(X, W, xw, nullptr, NN, COUT, 0);
    // per-node attention projections
    long long rn = (long long)RR * NN;
    k_qk<<<cdiv(rn * HH, 256), 256, 0, s>>>(xw, q, k, qn, kn, rn, COUT, HH);
    // init softmax state + output accumulator
    k_fill<<<cdiv((long long)NN * HH, 256), 256, 0, s>>>(amax, -INFINITY, (long long)NN * HH);
    k_fill<<<cdiv((long long)NN * HH, 256), 256, 0, s>>>(denom, 0.0f, (long long)NN * HH);
    k_init_bias<<<cdiv((long long)NN * COUT, 256), 256, 0, s>>>(outbuf, b, NN, COUT);
    // segment softmax over incoming edges
    long long eh = (long long)EE * HH;
    k_edge_alpha<<<cdiv(eh, 256), 256, 0, s>>>(qn, kn, src, dst, et, a_out, amax, EE, HH, NN);
    k_edge_exp<<<cdiv(eh, 256), 256, 0, s>>>(amax, dst, a_out, denom, EE, HH);
    k_edge_norm<<<cdiv(eh, 256), 256, 0, s>>>(denom, dst, a_out, EE, HH);
    // weighted message scatter
    k_edge_scatter<<<cdiv((long long)EE * COUT, 256), 256, 0, s>>>(
        xw, a_out, src, dst, et, outbuf, EE, COUT, OC1, HH, NN);
}

extern "C" void kernel_launch(void* const* d_in, const int* in_sizes, int n_in,
                              void* d_out, int out_size, void* d_ws, size_t ws_size,
                              hipStream_t stream) {
    const float* kg  = (const float*)d_in[0];
    const float* cc0 = (const float*)d_in[1];
    const float* w1  = (const float*)d_in[2];
    const float* b1  = (const float*)d_in[3];
    const float* w2  = (const float*)d_in[4];
    const float* b2  = (const float*)d_in[5];
    const float* r1w = (const float*)d_in[6];
    const float* r1q = (const float*)d_in[7];
    const float* r1k = (const float*)d_in[8];
    const float* r1b = (const float*)d_in[9];
    const float* r2w = (const float*)d_in[10];
    const float* r2q = (const float*)d_in[11];
    const float* r2k = (const float*)d_in[12];
    const float* r2b = (const float*)d_in[13];
    const float* lw  = (const float*)d_in[14];
    const float* lb  = (const float*)d_in[15];
    const int* eidx  = (const int*)d_in[16];
    const int* etype = (const int*)d_in[17];
    const int* src = eidx;
    const int* dst = eidx + EE;

    // ---- workspace layout (floats) ----
    float* ws = (float*)d_ws;
    size_t off = 0;
    float* deg  = ws + off; off += NN;                       // degree -> dis (in place)
    float* h1   = ws + off; off += (size_t)NN * 32;          // ccle0 @ W1
    float* g1   = ws + off; off += (size_t)NN * 32;          // gcn1 scattered
    float* h2   = ws + off; off += (size_t)NN * 128;         // leaky(g1) @ W2
    float* g2   = ws + off; off += (size_t)NN * 128;         // gcn2 scattered (= c)
    float* x0   = ws + off; off += (size_t)NN * 256;         // concat(kg, c)
    float* xw   = ws + off; off += (size_t)RR * NN * 256;    // per-relation projections
    float* qn   = ws + off; off += (size_t)RR * NN * HH;
    float* kn   = ws + off; off += (size_t)RR * NN * HH;
    float* amax = ws + off; off += (size_t)NN * HH;
    float* denom= ws + off; off += (size_t)NN * HH;
    float* hbuf = ws + off; off += (size_t)NN * 256;         // RGAT1 output

    float* out_final = (float*)d_out;                        // [N,256]
    float* a1 = out_final + (size_t)NN * 256;                // [E,4]
    float* a2 = a1 + (size_t)EE * HH;                        // [E,4]

    // ---- GCN normalization ----
    k_fill<<<cdiv(NN, 256), 256, 0, stream>>>(deg, 1.0f, NN);          // self loops
    k_deg_edges<<<cdiv(EE, 256), 256, 0, stream>>>(deg, dst, EE);
    k_rsqrt<<<cdiv(NN, 256), 256, 0, stream>>>(deg, NN);               // deg -> dis

    // ---- GCN layer 1 (4 -> 32) ----
    k_small_mm<<<cdiv((long long)NN * 32, 256), 256, 0, stream>>>(cc0, w1, h1, NN, 4, 32);
    k_gcn_self<<<cdiv((long long)NN * 32, 256), 256, 0, stream>>>(h1, deg, b1, g1, NN, 32);
    k_gcn_scatter<<<cdiv((long long)EE * 32, 256), 256, 0, stream>>>(h1, deg, src, dst, g1, EE, 32);
    k_leaky<<<cdiv((long long)NN * 32, 256), 256, 0, stream>>>(g1, SLOPE_ACT, (long long)NN * 32);

    // ---- GCN layer 2 (32 -> 128) ----
    k_small_mm<<<cdiv((long long)NN * 128, 256), 256, 0, stream>>>(g1, w2, h2, NN, 32, 128);
    k_gcn_self<<<cdiv((long long)NN * 128, 256), 256, 0, stream>>>(h2, deg, b2, g2, NN, 128);
    k_gcn_scatter<<<cdiv((long long)EE * 128, 256), 256, 0, stream>>>(h2, deg, src, dst, g2, EE, 128);

    // ---- x0 = concat(kg_emb, c) ----
    k_concat<<<cdiv((long long)NN * 256, 256), 256, 0, stream>>>(kg, g2, x0, NN);

    // ---- RGAT layer 1 ----
    rgat_layer(x0, r1w, r1q, r1k, r1b, xw, qn, kn, amax, denom, hbuf, a1,
               src, dst, etype, stream);
    k_leaky<<<cdiv((long long)NN * 256, 256), 256, 0, stream>>>(hbuf, SLOPE_ACT, (long long)NN * 256);

    // ---- RGAT layer 2 (writes into d_out's out region) ----
    rgat_layer(hbuf, r2w, r2q, r2k, r2b, xw, qn, kn, amax, denom, out_final, a2,
               src, dst, etype, stream);

    // ---- residual: out += x0 @ lin_w + lin_b  (WMMA, accumulate) ----
    dim3 gl(COUT / 64, NN / 16, 1);
    k_wmma_gemm<CIN><<<gl, 32, 0, stream>>>(x0, lw, out_final, lb, NN, COUT, 1);
}